// Attention_61778809586053
// MI455X (gfx1250) — compile-verified
//
#include <hip/hip_runtime.h>
#include <hip/hip_bf16.h>
#include <math.h>

typedef __attribute__((ext_vector_type(2))) float v2f;
typedef __attribute__((ext_vector_type(8))) float v8f;

#define BB 64
#define LL 1024
#define NN 1024

// ---------------------------------------------------------------------------
// Kernel 1: pre[b,m] = bias[m] + sum_n s_t[b,n] * W_s[m,n]
// One wave per 16x16 output tile, K=1024 via V_WMMA_F32_16X16X4_F32.
// ---------------------------------------------------------------------------
__global__ __launch_bounds__(32) void pre_kernel(const float* __restrict__ s_t,
                                                 const float* __restrict__ W_s,
                                                 const float* __restrict__ bias,
                                                 float* __restrict__ pre) {
  const int b0 = blockIdx.x * 16;
  const int m0 = blockIdx.y * 16;
  const int lane = threadIdx.x & 31;
  const int j  = lane & 15;   // A: M-row / B: N-col held by this lane
  const int hi = lane >> 4;   // K split: low half K={0,1}, high half K={2,3}
  const float* arow = s_t + (size_t)(b0 + j) * NN + 2 * hi;
  const float* brow = W_s + (size_t)(m0 + j) * NN + 2 * hi;
  v8f c = {0.f, 0.f, 0.f, 0.f, 0.f, 0.f, 0.f, 0.f};
#pragma unroll 8
  for (int k = 0; k < NN; k += 4) {
    v2f a  = *(const v2f*)(arow + k);
    v2f bm = *(const v2f*)(brow + k);
    c = __builtin_amdgcn_wmma_f32_16x16x4_f32(false, a, false, bm, (short)0, c,
                                              false, false);
  }
  const float bv = bias[m0 + j];
#pragma unroll
  for (int rl = 0; rl < 8; ++rl) {
    // D layout: VGPR rl holds row M = rl + 8*hi, col N = lane&15
    pre[(size_t)(b0 + rl + 8 * hi) * NN + (m0 + j)] = c[rl] + bv;
  }
}

// ---------------------------------------------------------------------------
// Kernel 2: scores[b,l] = sum_m tanh(h@W_h^T + pre[b,m] + cov*W_c[m]) * v[m]
// One block per 16-row tile of (b*L+l).  h tile staged in LDS once; 8 waves
// each own 8 m-tiles; feat never touches memory.
// ---------------------------------------------------------------------------
__global__ __launch_bounds__(256) void scores_kernel(const float* __restrict__ h,
                                                     const float* __restrict__ coverage,
                                                     const float* __restrict__ W_h,
                                                     const float* __restrict__ pre,
                                                     const float* __restrict__ W_c,
                                                     const float* __restrict__ vvec,
                                                     float* __restrict__ scores) {
  extern __shared__ float smem[];
  float* sh_h    = smem;                 // 16*1024 floats (64 KB)
  float* sh_cov  = smem + 16 * NN;       // 16 floats
  float* sh_part = sh_cov + 16;          // 8*16 floats

  const int r0 = blockIdx.x * 16;        // flat row = b*L + l (tile never crosses b)
  const int b  = r0 >> 10;               // r0 / L
  const int t  = threadIdx.x;

  // Cooperative load: 16 rows x 1024 cols of h, fully coalesced float4s.
  const float* hbase = h + (size_t)r0 * NN;
#pragma unroll
  for (int it = 0; it < 16; ++it) {
    const int idx = it * NN + t * 4;
    *(float4*)(sh_h + idx) = *(const float4*)(hbase + idx);
  }
  if (t < 16) sh_cov[t] = coverage[r0 + t];
  __syncthreads();

  const int w    = t >> 5;
  const int lane = t & 31;
  const int j    = lane & 15;
  const int hi   = lane >> 4;

  float covr[8];
#pragma unroll
  for (int rl = 0; rl < 8; ++rl) covr[rl] = sh_cov[rl + 8 * hi];

  const float* arow = sh_h + j * NN + 2 * hi;

  float vacc[8] = {0.f, 0.f, 0.f, 0.f, 0.f, 0.f, 0.f, 0.f};
  for (int mt = 0; mt < 8; ++mt) {
    const int m = (w * 8 + mt) * 16 + j;           // this lane's m column
    const float* brow = W_h + (size_t)m * NN + 2 * hi;  // W_h row (L2-resident)
    v8f c = {0.f, 0.f, 0.f, 0.f, 0.f, 0.f, 0.f, 0.f};
#pragma unroll 8
    for (int k = 0; k < NN; k += 4) {
      v2f a  = *(const v2f*)(arow + k);   // ds_load_b64 from staged tile
      v2f bm = *(const v2f*)(brow + k);   // global_load_b64, L2 hit
      c = __builtin_amdgcn_wmma_f32_16x16x4_f32(false, a, false, bm, (short)0,
                                                c, false, false);
    }
    // Fused epilogue on accumulator registers (feat never materialized).
    const float pm = pre[(size_t)b * NN + m];
    const float wc = W_c[m];
    const float vm = vvec[m];
#pragma unroll
    for (int rl = 0; rl < 8; ++rl) {
      const float f = tanhf(c[rl] + pm + covr[rl] * wc);
      vacc[rl] += f * vm;   // partial score for row rl + 8*hi
    }
  }

  // Reduce across the 16 lanes of each half (N-dim of the D tile).
#pragma unroll
  for (int rl = 0; rl < 8; ++rl) {
    float s = vacc[rl];
    s += __shfl_xor(s, 1, 32);
    s += __shfl_xor(s, 2, 32);
    s += __shfl_xor(s, 4, 32);
    s += __shfl_xor(s, 8, 32);
    vacc[rl] = s;
  }
  if (j == 0) {  // lanes 0 and 16 hold row sums for rows 0..7 / 8..15
#pragma unroll
    for (int rl = 0; rl < 8; ++rl) sh_part[w * 16 + rl + 8 * hi] = vacc[rl];
  }
  __syncthreads();
  if (t < 16) {  // deterministic cross-wave combine (no atomics)
    float s = 0.f;
#pragma unroll
    for (int w2 = 0; w2 < 8; ++w2) s += sh_part[w2 * 16 + t];
    scores[r0 + t] = s;
  }
}

// ---------------------------------------------------------------------------
// Kernel 3: per-batch softmax over L, attn_dist / coverage_new outputs, and
// context[b,n] = sum_l attn[l]*h[b,l,n] with coalesced row sweeps.
// ---------------------------------------------------------------------------
__global__ __launch_bounds__(256) void softmax_ctx_kernel(const float* __restrict__ h,
                                                          const float* __restrict__ coverage,
                                                          const float* __restrict__ scores,
                                                          float* __restrict__ attn_out,
                                                          float* __restrict__ ctx_out,
                                                          float* __restrict__ covnew_out) {
  __shared__ float attn_sh[LL];
  __shared__ float red[256];
  const int b = blockIdx.x;
  const int t = threadIdx.x;

  float sl[4];
#pragma unroll
  for (int q = 0; q < 4; ++q) sl[q] = scores[b * LL + t + q * 256];
  float mloc = fmaxf(fmaxf(sl[0], sl[1]), fmaxf(sl[2], sl[3]));
  red[t] = mloc;
  __syncthreads();
  for (int s = 128; s > 0; s >>= 1) {
    if (t < s) red[t] = fmaxf(red[t], red[t + s]);
    __syncthreads();
  }
  const float mx = red[0];
  __syncthreads();

  float ssum = 0.f;
#pragma unroll
  for (int q = 0; q < 4; ++q) {
    const float e = expf(sl[q] - mx);
    attn_sh[t + q * 256] = e;
    ssum += e;
  }
  red[t] = ssum;
  __syncthreads();
  for (int s = 128; s > 0; s >>= 1) {
    if (t < s) red[t] += red[t + s];
    __syncthreads();
  }
  const float inv = 1.0f / red[0];
  __syncthreads();

#pragma unroll
  for (int q = 0; q < 4; ++q) {
    const int l = t + q * 256;
    const float a = attn_sh[l] * inv;
    attn_sh[l] = a;
    attn_out[b * LL + l] = a;
    covnew_out[b * LL + l] = coverage[b * LL + l] + a;
  }
  __syncthreads();

  float acc[4] = {0.f, 0.f, 0.f, 0.f};
  const float* hb = h + (size_t)b * LL * NN;
  for (int l = 0; l < LL; ++l) {
    const float a = attn_sh[l];
    const float* hr = hb + (size_t)l * NN;
#pragma unroll
    for (int q = 0; q < 4; ++q) acc[q] = fmaf(a, hr[t + q * 256], acc[q]);
  }
#pragma unroll
  for (int q = 0; q < 4; ++q) ctx_out[b * NN + t + q * 256] = acc[q];
}

// ---------------------------------------------------------------------------
extern "C" void kernel_launch(void* const* d_in, const int* in_sizes, int n_in,
                              void* d_out, int out_size, void* d_ws, size_t ws_size,
                              hipStream_t stream) {
  (void)in_sizes; (void)n_in; (void)out_size; (void)ws_size;
  const float* h        = (const float*)d_in[0];  // [B,L,N]
  const float* s_t      = (const float*)d_in[1];  // [B,N]
  const float* coverage = (const float*)d_in[2];  // [B,L]
  const float* W_h      = (const float*)d_in[3];  // [N,N]
  const float* W_s      = (const float*)d_in[4];  // [N,N]
  const float* W_c      = (const float*)d_in[5];  // [N,1]
  const float* vvec     = (const float*)d_in[6];  // [1,N]
  const float* bias     = (const float*)d_in[7];  // [N]

  float* out     = (float*)d_out;
  float* attn    = out;                       // [B,L]
  float* ctx     = out + BB * LL;             // [B,N]
  float* covnew  = out + BB * LL + BB * NN;   // [B,L]

  float* pre    = (float*)d_ws;               // [B,N]  = 256 KB
  float* scores = pre + (size_t)BB * NN;      // [B,L]  = 256 KB

  pre_kernel<<<dim3(BB / 16, NN / 16), 32, 0, stream>>>(s_t, W_s, bias, pre);

  const size_t smem_bytes = (size_t)(16 * NN + 16 + 8 * 16) * sizeof(float); // 66112 B
  scores_kernel<<<(BB * LL) / 16, 256, smem_bytes, stream>>>(h, coverage, W_h, pre,
                                                             W_c, vvec, scores);

  softmax_ctx_kernel<<<BB, 256, 0, stream>>>(h, coverage, scores, attn, ctx, covnew);
}